// FrameWork_67345087201450
// MI455X (gfx1250) — compile-verified
//
#include <hip/hip_runtime.h>
#include <hip/hip_bf16.h>

typedef float v2f __attribute__((ext_vector_type(2)));
typedef float v8f __attribute__((ext_vector_type(8)));

#define D 64
#define NROWS 400000   // B * N_ENT
#define N_REL 401
#define N_B   8

// ---------------------------------------------------------------------------
// Zero-init workspace regions used accumulatively (agg + touched)
// ---------------------------------------------------------------------------
__global__ void zero_ws_kernel(float* __restrict__ agg, int* __restrict__ touched,
                               int nAgg, int nT) {
    int i = blockIdx.x * blockDim.x + threadIdx.x;
    int stride = gridDim.x * blockDim.x;
    for (int j = i; j < nAgg; j += stride) agg[j] = 0.0f;
    for (int j = i; j < nT;   j += stride) touched[j] = 0;
}

// ---------------------------------------------------------------------------
// Tiny precomputes:
//   Qq[b][a]    = sum_d query[b][d]   * Wqr_w[a][d] + Wqr_b[a]     (8 x 64)
//   Rr[r][a]    = sum_d rela[r][d]    * Wr[a][d]                   (401 x 64)
//   Wfold[i][j] = sum_k W2[i][k] * W1[k][j]                        (64 x 64)
//   bfold[i]    = sum_k W2[i][k] * b1[k] + b2[i]                   (64)
// ---------------------------------------------------------------------------
__global__ void prep_kernel(const float* __restrict__ query,
                            const float* __restrict__ Wqr_w,
                            const float* __restrict__ Wqr_b,
                            const float* __restrict__ rela,
                            const float* __restrict__ Wr,
                            const float* __restrict__ w1,
                            const float* __restrict__ b1,
                            const float* __restrict__ w2,
                            const float* __restrict__ b2,
                            float* __restrict__ Qq,
                            float* __restrict__ Rr,
                            float* __restrict__ Wfold,
                            float* __restrict__ bfold) {
    const int QQ_N = N_B * D;          // 512
    const int RR_N = N_REL * D;        // 25664
    const int WF_N = D * D;            // 4096
    int i = blockIdx.x * blockDim.x + threadIdx.x;
    if (i < QQ_N) {
        int b = i >> 6, a = i & 63;
        float s = Wqr_b[a];
        #pragma unroll 8
        for (int d = 0; d < D; ++d) s += query[b * D + d] * Wqr_w[a * D + d];
        Qq[i] = s;
    } else if (i < QQ_N + RR_N) {
        int j = i - QQ_N;
        int r = j >> 6, a = j & 63;
        float s = 0.0f;
        #pragma unroll 8
        for (int d = 0; d < D; ++d) s += rela[r * D + d] * Wr[a * D + d];
        Rr[j] = s;
    } else if (i < QQ_N + RR_N + WF_N) {
        int j = i - QQ_N - RR_N;
        int ii = j >> 6, jj = j & 63;
        float s = 0.0f;
        #pragma unroll 8
        for (int k = 0; k < D; ++k) s += w2[ii * D + k] * w1[k * D + jj];
        Wfold[j] = s;
    } else if (i < QQ_N + RR_N + WF_N + D) {
        int ii = i - QQ_N - RR_N - WF_N;
        float s = b2[ii];
        #pragma unroll 8
        for (int k = 0; k < D; ++k) s += w2[ii * D + k] * b1[k];
        bfold[ii] = s;
    }
}

// ---------------------------------------------------------------------------
// WMMA helper: one wave computes a 16-row strip of  OUT = X @ W.T
// (X: [rows x 64] f32, W: [64 x 64] f32), using V_WMMA_F32_16X16X4_F32.
// A 16x4 layout (ISA 7.12.2): lanes 0-15 hold K={0,1}, lanes 16-31 K={2,3}.
// B 4x16 is the symmetric transpose layout. C/D: 8 VGPRs, lane&15 = N,
// row = r + 8*(lane>>4).
// ---------------------------------------------------------------------------
__device__ __forceinline__ void gemm16_strip(const float* __restrict__ X,
                                             const float* __restrict__ W,
                                             int row0, v8f acc[4]) {
    const int lane = threadIdx.x & 31;
    const int m  = lane & 15;     // M (for A) / N (for B) within tile
    const int kh = lane >> 4;     // K-half selector
    #pragma unroll
    for (int ks = 0; ks < 16; ++ks) {
        const int k0 = ks * 4 + 2 * kh;
        v2f a;
        a.x = X[(row0 + m) * D + k0];
        a.y = X[(row0 + m) * D + k0 + 1];
        #pragma unroll
        for (int nt = 0; nt < 4; ++nt) {
            const int n = nt * 16 + m;
            v2f b;
            b.x = W[n * D + k0];
            b.y = W[n * D + k0 + 1];
            acc[nt] = __builtin_amdgcn_wmma_f32_16x16x4_f32(
                false, a, false, b, (short)0, acc[nt], false, false);
        }
    }
}

// ---------------------------------------------------------------------------
// Hs = hidden @ Ws.T   (400000 x 64) — one 16-row strip per wave
// ---------------------------------------------------------------------------
__global__ __launch_bounds__(256) void hs_gemm_kernel(const float* __restrict__ hidden,
                                                      const float* __restrict__ Ws,
                                                      float* __restrict__ Hs) {
    const int wave = blockIdx.x * (blockDim.x >> 5) + (threadIdx.x >> 5);
    const int row0 = wave * 16;
    if (row0 >= NROWS) return;
    v8f acc[4];
    #pragma unroll
    for (int nt = 0; nt < 4; ++nt)
        #pragma unroll
        for (int r = 0; r < 8; ++r) acc[nt][r] = 0.0f;
    gemm16_strip(hidden, Ws, row0, acc);
    const int lane = threadIdx.x & 31;
    const int m = lane & 15, kh = lane >> 4;
    #pragma unroll
    for (int r = 0; r < 8; ++r) {
        const int row = row0 + r + 8 * kh;
        #pragma unroll
        for (int nt = 0; nt < 4; ++nt)
            Hs[row * D + nt * 16 + m] = acc[nt][r];
    }
}

// ---------------------------------------------------------------------------
// Edge phase: one edge per wave32. lanes cover dims {lane, lane+32}.
//   alpha = sigmoid( wattn . relu(Hs[s] + Rr[r] + Qq[b]) )
//   agg[obj] += hidden[s] * rela[r] * alpha   (f32 global atomics)
// ---------------------------------------------------------------------------
__global__ __launch_bounds__(256) void edge_kernel(const float* __restrict__ hidden,
                                                   const float* __restrict__ rela,
                                                   const float* __restrict__ Hs,
                                                   const float* __restrict__ Rr,
                                                   const float* __restrict__ Qq,
                                                   const float* __restrict__ wattn,
                                                   const int* __restrict__ id_bat,
                                                   const int* __restrict__ id_sub,
                                                   const int* __restrict__ id_rel,
                                                   const int* __restrict__ id_obj,
                                                   float* __restrict__ agg,
                                                   int* __restrict__ touched,
                                                   int E) {
    const int lane = threadIdx.x & 31;
    const int wid = blockIdx.x * (blockDim.x >> 5) + (threadIdx.x >> 5);
    const int nw  = gridDim.x * (blockDim.x >> 5);
    const float wa0 = wattn[lane];
    const float wa1 = wattn[lane + 32];
    for (int e = wid; e < E; e += nw) {
        const int s  = id_sub[e];
        const int r  = id_rel[e];
        const int bq = id_bat[e];
        const int o  = id_obj[e];
        float pre0 = Hs[s * D + lane]      + Rr[r * D + lane]      + Qq[bq * D + lane];
        float pre1 = Hs[s * D + lane + 32] + Rr[r * D + lane + 32] + Qq[bq * D + lane + 32];
        pre0 = fmaxf(pre0, 0.0f);
        pre1 = fmaxf(pre1, 0.0f);
        float t = pre0 * wa0 + pre1 * wa1;
        #pragma unroll
        for (int mask = 16; mask > 0; mask >>= 1) t += __shfl_xor(t, mask, 32);
        const float alpha = 1.0f / (1.0f + __expf(-t));
        const float m0 = hidden[s * D + lane]      * rela[r * D + lane]      * alpha;
        const float m1 = hidden[s * D + lane + 32] * rela[r * D + lane + 32] * alpha;
        __hip_atomic_fetch_add(&agg[o * D + lane],      m0,
                               __ATOMIC_RELAXED, __HIP_MEMORY_SCOPE_AGENT);
        __hip_atomic_fetch_add(&agg[o * D + lane + 32], m1,
                               __ATOMIC_RELAXED, __HIP_MEMORY_SCOPE_AGENT);
        if (lane == 0) touched[o] = 1;
    }
}

// ---------------------------------------------------------------------------
// out = touched ? relu(agg @ Wfold.T + bfold) : 0    (WMMA, masked store)
// ---------------------------------------------------------------------------
__global__ __launch_bounds__(256) void final_gemm_kernel(const float* __restrict__ agg,
                                                         const float* __restrict__ Wfold,
                                                         const float* __restrict__ bfold,
                                                         const int* __restrict__ touched,
                                                         float* __restrict__ out) {
    const int wave = blockIdx.x * (blockDim.x >> 5) + (threadIdx.x >> 5);
    const int row0 = wave * 16;
    if (row0 >= NROWS) return;
    v8f acc[4];
    #pragma unroll
    for (int nt = 0; nt < 4; ++nt)
        #pragma unroll
        for (int r = 0; r < 8; ++r) acc[nt][r] = 0.0f;
    gemm16_strip(agg, Wfold, row0, acc);
    const int lane = threadIdx.x & 31;
    const int m = lane & 15, kh = lane >> 4;
    float bias[4];
    #pragma unroll
    for (int nt = 0; nt < 4; ++nt) bias[nt] = bfold[nt * 16 + m];
    #pragma unroll
    for (int r = 0; r < 8; ++r) {
        const int row = row0 + r + 8 * kh;
        const int tch = touched[row];
        #pragma unroll
        for (int nt = 0; nt < 4; ++nt) {
            const float v = tch ? fmaxf(acc[nt][r] + bias[nt], 0.0f) : 0.0f;
            out[row * D + nt * 16 + m] = v;
        }
    }
}

// ---------------------------------------------------------------------------
// Launch
// ---------------------------------------------------------------------------
extern "C" void kernel_launch(void* const* d_in, const int* in_sizes, int n_in,
                              void* d_out, int out_size, void* d_ws, size_t ws_size,
                              hipStream_t stream) {
    const float* query   = (const float*)d_in[0];
    const float* hidden  = (const float*)d_in[1];
    const float* Ws      = (const float*)d_in[2];
    const float* Wr      = (const float*)d_in[3];
    const float* Wqr_w   = (const float*)d_in[4];
    const float* Wqr_b   = (const float*)d_in[5];
    const float* Wattn   = (const float*)d_in[6];
    const float* rela    = (const float*)d_in[7];
    const float* mlp_w1  = (const float*)d_in[8];
    const float* mlp_b1  = (const float*)d_in[9];
    const float* mlp_w2  = (const float*)d_in[10];
    const float* mlp_b2  = (const float*)d_in[11];
    const int*   id_bat  = (const int*)d_in[12];
    const int*   id_sub  = (const int*)d_in[13];
    const int*   id_rel  = (const int*)d_in[14];
    const int*   id_obj  = (const int*)d_in[15];
    const int E = in_sizes[12];

    // workspace carve-up (f32 words unless noted)
    char* base = (char*)d_ws;
    const size_t AGG_N = (size_t)NROWS * D;           // 25,600,000
    float* agg   = (float*)base;                              base += AGG_N * 4;
    float* Hs    = (float*)base;                              base += AGG_N * 4;
    float* Rr    = (float*)base;                              base += (size_t)N_REL * D * 4;
    float* Qq    = (float*)base;                              base += (size_t)N_B * D * 4;
    float* Wfold = (float*)base;                              base += (size_t)D * D * 4;
    float* bfold = (float*)base;                              base += 256;
    int*   touched = (int*)base;

    // 1) zero accumulators
    zero_ws_kernel<<<2048, 256, 0, stream>>>(agg, touched, (int)AGG_N, NROWS);

    // 2) tiny precomputes (Qq, Rr, Wfold, bfold)
    const int PREP_N = N_B * D + N_REL * D + D * D + D;       // 30,336
    prep_kernel<<<(PREP_N + 255) / 256, 256, 0, stream>>>(
        query, Wqr_w, Wqr_b, rela, Wr, mlp_w1, mlp_b1, mlp_w2, mlp_b2,
        Qq, Rr, Wfold, bfold);

    // 3) Hs = hidden @ Ws.T  (WMMA): 25000 waves, 8 waves/block
    hs_gemm_kernel<<<(NROWS / 16 + 7) / 8, 256, 0, stream>>>(hidden, Ws, Hs);

    // 4) edge gather / gate / scatter-add
    edge_kernel<<<8192, 256, 0, stream>>>(hidden, rela, Hs, Rr, Qq, Wattn,
                                          id_bat, id_sub, id_rel, id_obj,
                                          agg, touched, E);

    // 5) masked folded-MLP (WMMA) -> d_out
    final_gemm_kernel<<<(NROWS / 16 + 7) / 8, 256, 0, stream>>>(
        agg, Wfold, bfold, touched, (float*)d_out);
}